// VolumeRenderer_5703716569543
// MI455X (gfx1250) — compile-verified
//
#include <hip/hip_runtime.h>
#include <stdint.h>

// Volume rendering weighted sum (R=131072 rays x N=128 samples):
//   trans = exclusive cumprod(1 - alpha + 1e-10) along N
//   out[r,c] = sum_n trans[r,n]*alpha[r,n]*rgbs[r,n,c]
//
// Memory-bound (268 MB in, 1.5 MB out, ~0.56 FLOP/byte) -> design for 23.3 TB/s.
// One wave32 per ray; CDNA5 async global->LDS (B128, ASYNCcnt) double-buffered
// staging gives perfectly coalesced global traffic and hides latency; the
// 16B-coalesced -> 48B/lane rgb swizzle happens in LDS for free.

#define WAVES_PER_BLOCK 8
#define BLOCK_THREADS   (WAVES_PER_BLOCK * 32)
#define RAYS_PER_WAVE   4
#define N_SAMPLES       128

__device__ __forceinline__ unsigned lds_off(const void* p) {
  // generic LDS pointer = {shared-aperture hi bits, 32-bit LDS offset}
  return (unsigned)(uintptr_t)p;
}

// Stage one ray (512B alpha + 1536B rgb) into this wave's LDS buffer.
// Per the ISA, the instruction offset is added to BOTH the global and LDS
// addresses, so one (lds,addr) pair covers the three rgb B128 chunks.
__device__ __forceinline__ void async_prefetch_ray(const char* ga, const char* gr,
                                                   unsigned ldsA, unsigned ldsR) {
  asm volatile("global_load_async_to_lds_b128 %0, %1, off"
               :: "v"(ldsA), "v"(ga) : "memory");
  asm volatile("global_load_async_to_lds_b128 %0, %1, off"
               :: "v"(ldsR), "v"(gr) : "memory");
  asm volatile("global_load_async_to_lds_b128 %0, %1, off offset:512"
               :: "v"(ldsR), "v"(gr) : "memory");
  asm volatile("global_load_async_to_lds_b128 %0, %1, off offset:1024"
               :: "v"(ldsR), "v"(gr) : "memory");
}

__global__ __launch_bounds__(BLOCK_THREADS)
void volume_render_kernel(const float* __restrict__ alpha,
                          const float* __restrict__ rgbs,
                          float* __restrict__ out, int R) {
  // 8 waves * 2 buffers * 2048 B = 32 KB LDS (WGP has 320 KB)
  __shared__ float4 smem[WAVES_PER_BLOCK * 2 * 128];

  const int lane   = threadIdx.x & 31;
  const int wave   = threadIdx.x >> 5;
  const int gwave  = blockIdx.x * WAVES_PER_BLOCK + wave;
  const int nwaves = gridDim.x * WAVES_PER_BLOCK;

  float4* const buf0 = &smem[(wave * 2 + 0) * 128];
  float4* const buf1 = &smem[(wave * 2 + 1) * 128];
  const unsigned base0 = lds_off(buf0);
  const unsigned base1 = lds_off(buf1);

  const char* const abytes = (const char*)alpha;
  const char* const rbytes = (const char*)rgbs;

  int ray = gwave;
  int cur = 0;
  if (ray < R) {
    async_prefetch_ray(abytes + (size_t)ray * 512  + lane * 16,
                       rbytes + (size_t)ray * 1536 + lane * 16,
                       base0 + lane * 16, base0 + 512 + lane * 16);
  }

  for (; ray < R; ray += nwaves) {
    const int nxt = ray + nwaves;
    if (nxt < R) {
      const unsigned nb = (cur == 0) ? base1 : base0;
      // close WAR on the buffer we read last iteration before rewriting it
      asm volatile("s_wait_dscnt 0" ::: "memory");
      async_prefetch_ray(abytes + (size_t)nxt * 512  + lane * 16,
                         rbytes + (size_t)nxt * 1536 + lane * 16,
                         nb + lane * 16, nb + 512 + lane * 16);
      asm volatile("s_wait_asynccnt 4" ::: "memory");  // current buffer's 4 ops done
    } else {
      asm volatile("s_wait_asynccnt 0" ::: "memory");
    }

    const float4* b = (cur == 0) ? buf0 : buf1;

    // Lane i owns samples 4i..4i+3.
    const float4 a4 = b[lane];
    const float4 c0 = b[32 + lane * 3 + 0];  // {r0,g0,b0,r1}
    const float4 c1 = b[32 + lane * 3 + 1];  // {g1,b1,r2,g2}
    const float4 c2 = b[32 + lane * 3 + 2];  // {b2,r3,g3,b3}

    const float EPS = 1e-10f;
    const float q0 = 1.0f - a4.x + EPS;
    const float q1 = 1.0f - a4.y + EPS;
    const float q2 = 1.0f - a4.z + EPS;
    const float q3 = 1.0f - a4.w + EPS;

    // Wave-level exclusive product scan (wave32: one wave == one ray's scan).
    float s = q0 * q1 * q2 * q3;
#pragma unroll
    for (int d = 1; d < 32; d <<= 1) {
      const float o = __shfl_up(s, d, 32);
      if (lane >= d) s *= o;
    }
    float e = __shfl_up(s, 1, 32);   // product of all samples before lane's chunk
    if (lane == 0) e = 1.0f;

    const float t0 = e;
    const float t1 = t0 * q0;
    const float t2 = t1 * q1;
    const float t3 = t2 * q2;
    const float w0 = t0 * a4.x;
    const float w1 = t1 * a4.y;
    const float w2 = t2 * a4.z;
    const float w3 = t3 * a4.w;

    float r = w0 * c0.x + w1 * c0.w + w2 * c1.z + w3 * c2.y;
    float g = w0 * c0.y + w1 * c1.x + w2 * c1.w + w3 * c2.z;
    float bb = w0 * c0.z + w1 * c1.y + w2 * c2.x + w3 * c2.w;

#pragma unroll
    for (int d = 16; d >= 1; d >>= 1) {
      r  += __shfl_xor(r,  d, 32);
      g  += __shfl_xor(g,  d, 32);
      bb += __shfl_xor(bb, d, 32);
    }

    if (lane == 0) {
      float* o3 = out + (size_t)ray * 3;
      o3[0] = r; o3[1] = g; o3[2] = bb;
    }
    cur ^= 1;
  }
}

extern "C" void kernel_launch(void* const* d_in, const int* in_sizes, int n_in,
                              void* d_out, int out_size, void* d_ws, size_t ws_size,
                              hipStream_t stream) {
  (void)n_in; (void)out_size; (void)d_ws; (void)ws_size;
  const float* alpha = (const float*)d_in[0];
  const float* rgbs  = (const float*)d_in[1];
  float* out = (float*)d_out;

  const int R = in_sizes[0] / N_SAMPLES;
  int blocks = (R + WAVES_PER_BLOCK * RAYS_PER_WAVE - 1) / (WAVES_PER_BLOCK * RAYS_PER_WAVE);
  if (blocks < 1) blocks = 1;

  volume_render_kernel<<<blocks, BLOCK_THREADS, 0, stream>>>(alpha, rgbs, out, R);
}